// SelfCorrectingPointProcess_13984413516461
// MI455X (gfx1250) — compile-verified
//
#include <hip/hip_runtime.h>
#include <math.h>

typedef __attribute__((ext_vector_type(2))) float v2f;
typedef __attribute__((ext_vector_type(8))) float v8f;

constexpr int T_DIM   = 4096;
constexpr int THREADS = 256;
constexpr int WAVES   = THREADS / 32;
constexpr int VEC_PER_THREAD = T_DIM / (THREADS * 4); // 4 float4 iters per thread

#define LOG2E_F 1.442695040888963f   // log2(e)
#define LN2_F   0.693147180559945f   // ln(2)

// Raw v_exp_f32: no denormal range-guard (args here are either benign or
// decay to 0, matching the reference's underflow-to-zero behavior).
__device__ __forceinline__ float fast_exp2(float x) {
    return __builtin_amdgcn_exp2f(x);
}

__device__ __forceinline__ float softplus_f(float x) {
    // numerically stable: max(x,0) + log1p(exp(-|x|))
    return fmaxf(x, 0.f) + log1pf(__expf(-fabsf(x)));
}

// Wave32 sum-reduction using V_WMMA_F32_16X16X4_F32.
// A (16x4): lane L holds A[L%16, {0,1}+2*(L/16)] in {a.x,a.y}. With B = all-ones,
// D[m, n] = sum_k A[m,k] = part(lane m) + part(lane m+16)   (a.y = 0).
// D layout: c[r] on lanes 0-15 = row r, on lanes 16-31 = row r+8.
// Sum c[0..7] -> lanes 0-15 hold rows 0-7 total, lanes 16-31 rows 8-15 total;
// one xor-16 shuffle gives the full 32-lane sum on every lane.
__device__ __forceinline__ float wave_reduce_wmma(float v) {
    v2f a; a.x = v;   a.y = 0.f;
    v2f b; b.x = 1.f; b.y = 1.f;
    v8f c = {};
    c = __builtin_amdgcn_wmma_f32_16x16x4_f32(
            /*neg_a=*/false, a, /*neg_b=*/false, b,
            /*c_mod=*/(short)0, c, /*reuse_a=*/false, /*reuse_b=*/false);
    float s = ((c[0] + c[1]) + (c[2] + c[3])) + ((c[4] + c[5]) + (c[6] + c[7]));
    s += __shfl_xor(s, 16, 32);
    return s;
}

__global__ __launch_bounds__(THREADS)
void scpp_kernel(const float* __restrict__ t_all,
                 const float* __restrict__ mu_p,
                 const float* __restrict__ beta_p,
                 float* __restrict__ out0,       // (N,)  loglik - compensator
                 float* __restrict__ lamb_out)   // (N,T) exp(loglik_mat)
{
    const int row = blockIdx.x;
    const float*  trow  = t_all + (size_t)row * T_DIM;
    const float4* trow4 = (const float4*)trow;
    float4*       lrow4 = (float4*)(lamb_out + (size_t)row * T_DIM);

    const float mu_s   = softplus_f(mu_p[0]);
    const float beta_s = softplus_f(beta_p[0]);
    const float inv_mu = 1.f / mu_s;

    // log2-domain constants: v_exp_f32 is base-2, so pre-scale by log2(e)
    // once per block instead of multiplying every element.
    const float inv_mu2 = inv_mu * LOG2E_F;           // log2e / mu
    const float beta2   = beta_s * LOG2E_F;           // beta * log2e
    const float mu2     = mu_s   * LOG2E_F;           // mu   * log2e
    const float c1      = 1.f - fast_exp2(-beta2);    // (1 - e^{-beta})

    const float t0    = trow[0];
    const float tlast = trow[T_DIM - 1];
    const float dt    = (tlast - t0) * (1.f / (float)(T_DIM - 1));

    float ll_acc2 = 0.f;  // sum of loglik_mat, scaled by log2e (undo at end)
    float cp_acc  = 0.f;  // sum of e_j * coef_j  (compensator * mu)

    const int tid = threadIdx.x;
#pragma unroll
    for (int it = 0; it < VEC_PER_THREAD; ++it) {
        const int vidx = it * THREADS + tid;
        const int j0   = vidx * 4;
        const float4 tv = trow4[vidx];
        const float tj[4] = {tv.x, tv.y, tv.z, tv.w};
        float lo[4];
#pragma unroll
        for (int k = 0; k < 4; ++k) {
            const int   j = j0 + k;
            const float t = tj[k];
            // log2-domain log-intensity: (t/mu - beta*grid) * log2e
            const float grid = fmaf(dt, (float)j, t0);
            const float llm2 = fmaf(-beta2, grid, t * inv_mu2);
            lo[k]   = fast_exp2(llm2);          // single v_exp_f32
            ll_acc2 += llm2;
            // compensator term: e^{mu t_j} * coef_j, fully fused in base-2
            float term;
            if (j == 0) {
                term = -fast_exp2(mu2 * t);                                   // coef = -1
            } else if (j == T_DIM - 1) {
                term = fast_exp2(fmaf(mu2, t, -beta2 * (float)(T_DIM - 2)));  // coef = w[T-2]
            } else {
                term = fast_exp2(fmaf(mu2, t, -beta2 * (float)(j - 1))) * c1;
            }
            cp_acc += term;
        }
        float4 lv; lv.x = lo[0]; lv.y = lo[1]; lv.z = lo[2]; lv.w = lo[3];
        lrow4[vidx] = lv;   // streaming B128 store
    }

    // Wave-level reductions through the matrix pipe (EXEC all-1s here).
    const float ll_w = wave_reduce_wmma(ll_acc2);
    const float cp_w = wave_reduce_wmma(cp_acc);

    __shared__ float s_ll[WAVES];
    __shared__ float s_cp[WAVES];
    const int wid = tid >> 5;
    if ((tid & 31) == 0) { s_ll[wid] = ll_w; s_cp[wid] = cp_w; }
    __syncthreads();
    if (tid == 0) {
        float L2 = 0.f, C = 0.f;
#pragma unroll
        for (int w = 0; w < WAVES; ++w) { L2 += s_ll[w]; C += s_cp[w]; }
        out0[row] = L2 * LN2_F - C * inv_mu;   // undo log2e scale on loglik
    }
}

extern "C" void kernel_launch(void* const* d_in, const int* in_sizes, int n_in,
                              void* d_out, int out_size, void* d_ws, size_t ws_size,
                              hipStream_t stream) {
    const float* t    = (const float*)d_in[0];   // (N, T, 1) fp32
    const float* mu   = (const float*)d_in[1];   // (1,) fp32
    const float* beta = (const float*)d_in[2];   // (1,) fp32

    const int N = in_sizes[0] / T_DIM;           // 4096

    float* out0 = (float*)d_out;                 // first N floats
    float* lamb = out0 + N;                      // then N*T floats

    scpp_kernel<<<N, THREADS, 0, stream>>>(t, mu, beta, out0, lamb);
}